// Aggregator_83691732730318
// MI455X (gfx1250) — compile-verified
//
#include <hip/hip_runtime.h>

typedef __attribute__((ext_vector_type(2))) float v2f;
typedef __attribute__((ext_vector_type(8))) float v8f;

#define DD    64   // feature dim
#define NN    32   // neighbors
#define TILE  16   // pairs (rows) per workgroup
#define LDSTR 68   // padded LDS row stride (floats): bank-conflict-free WMMA fragment loads

#if defined(__HIP_DEVICE_COMPILE__) && \
    __has_builtin(__builtin_amdgcn_global_load_async_to_lds_b128)
#define HAVE_ASYNC_LDS 1
typedef int v4i __attribute__((vector_size(16)));
typedef v4i __attribute__((address_space(1)))* gv4i_p;   // global 16B chunk
typedef v4i __attribute__((address_space(3)))* lv4i_p;   // LDS    16B chunk
#else
#define HAVE_ASYNC_LDS 0
#endif

__global__ __launch_bounds__(512) void Aggregator_83691732730318_kernel(
    const float* __restrict__ selfv,   // [B*E, 64]
    const float* __restrict__ nbrv,    // [B*E, 32, 64]
    const float* __restrict__ nbrr,    // [B*E, 32, 64]
    const float* __restrict__ uemb,    // [B, 64]
    const float* __restrict__ W,       // [64, 64] (out, in)
    float* __restrict__ out)           // [B*E, 64]
{
    __shared__ float uLds[DD];
    __shared__ float wLds[DD * LDSTR];
    __shared__ float xLds[TILE * LDSTR];

    const int  tid  = threadIdx.x;
    const int  lane = tid & 31;
    const int  wave = tid >> 5;                    // 0..15, one pair per wave
    const int  tile = blockIdx.x;                  // 0..2047
    const long pair = (long)tile * TILE + wave;    // flat b*E + e
    const long b    = pair >> 6;                   // E == 64, and 16 | 64 -> b uniform per block

    // ---- Stage W (row-padded) into LDS.
#if HAVE_ASYNC_LDS
    // Async DMA: 64x64 floats = 1024 x 16B chunks, 16 chunks per row; chunks
    // never cross the pad. Tracked by ASYNCcnt, waited before the barrier.
    for (int c = tid; c < (DD * DD) / 4; c += 512) {
        const int row = c >> 4;
        const int col = (c & 15) << 2;
        __builtin_amdgcn_global_load_async_to_lds_b128(
            (gv4i_p)(W + row * DD + col),
            (lv4i_p)&wLds[row * LDSTR + col],
            /*imm offset=*/0, /*cpol=*/0);
    }
#else
    for (int i = tid; i < DD * DD; i += 512)
        wLds[(i >> 6) * LDSTR + (i & 63)] = W[i];
#endif
    if (tid < DD) uLds[tid] = uemb[b * DD + tid];
#if HAVE_ASYNC_LDS
    asm volatile("s_wait_asynccnt 0x0" ::: "memory");
#endif
    __syncthreads();

    // ---- Phase 1: attention scores. Lane n handles neighbor n of this wave's pair.
    const float* __restrict__ relRow = nbrr + (pair * NN + lane) * DD;
    const float* __restrict__ vbase  = nbrv + pair * NN * DD;
    __builtin_prefetch(vbase + (size_t)lane * DD, 0, 0);   // warm neighbor_vectors tile

    float s = 0.f;
    #pragma unroll
    for (int d = 0; d < DD; d += 4) {
        float4 r4 = *(const float4*)(relRow + d);
        s += r4.x * uLds[d]     + r4.y * uLds[d + 1]
           + r4.z * uLds[d + 2] + r4.w * uLds[d + 3];
    }

    // Wave32 softmax over the 32 neighbor scores (one per lane).
    float mx = s;
    #pragma unroll
    for (int off = 16; off > 0; off >>= 1) mx = fmaxf(mx, __shfl_xor(mx, off, 32));
    float p = __expf(s - mx);
    float sm = p;
    #pragma unroll
    for (int off = 16; off > 0; off >>= 1) sm += __shfl_xor(sm, off, 32);
    float attn = p / sm;

    // ---- Phase 2: weighted aggregation; lane owns dims {lane, lane+32} (coalesced streams).
    float acc0 = 0.f, acc1 = 0.f;
    #pragma unroll 4
    for (int n = 0; n < NN; ++n) {
        float a = __shfl(attn, n, 32);             // broadcast lane n's weight
        acc0 += a * vbase[n * DD + lane];
        acc1 += a * vbase[n * DD + lane + 32];
    }
    float x0 = selfv[pair * DD + lane]      + acc0;
    float x1 = selfv[pair * DD + lane + 32] + acc1;
    xLds[wave * LDSTR + lane]      = x0;
    xLds[wave * LDSTR + lane + 32] = x1;
    __syncthreads();

    // ---- Phase 3: out = relu(X(16x64) @ W^T) via V_WMMA_F32_16X16X4_F32.
    // Waves 0..3 each own a 16-wide N tile (j0 = 16*wave). EXEC is all-ones per wave.
    if (wave < 4) {
        const int j0   = wave << 4;
        const int half = lane >> 4;                // 0: K={0,1}; 1: K={2,3}
        const int mn   = lane & 15;                // M (A) / N (B) index
        const int kb   = half << 1;
        v8f c = {};
        #pragma unroll
        for (int k0 = 0; k0 < DD; k0 += 4) {
            // A fragment: X[mn, k0+kb .. k0+kb+1]   (16x4 fp32 A layout, ISA 7.12.2)
            v2f av = *(const v2f*)&xLds[mn * LDSTR + k0 + kb];
            // B fragment: W^T[k0+kb.., j0+mn] == W[j0+mn, k0+kb..]
            v2f bv = *(const v2f*)&wLds[(j0 + mn) * LDSTR + k0 + kb];
            c = __builtin_amdgcn_wmma_f32_16x16x4_f32(
                    /*neg_a=*/false, av, /*neg_b=*/false, bv,
                    /*c_mod=*/(short)0, c, /*reuse_a=*/false, /*reuse_b=*/false);
        }
        // C/D layout: VGPR v -> row v (lanes 0-15) / row v+8 (lanes 16-31), col = lane&15.
        #pragma unroll
        for (int v = 0; v < 8; ++v) {
            int  row = v + (half << 3);
            long gi  = ((long)tile * TILE + row) * DD + j0 + mn;
            out[gi]  = fmaxf(c[v], 0.f);
        }
    }
}

extern "C" void kernel_launch(void* const* d_in, const int* in_sizes, int n_in,
                              void* d_out, int out_size, void* d_ws, size_t ws_size,
                              hipStream_t stream) {
    (void)in_sizes; (void)n_in; (void)d_ws; (void)ws_size; (void)out_size;
    const float* selfv = (const float*)d_in[0];   // [512,64,64]
    const float* nbrv  = (const float*)d_in[1];   // [512,64,32,64]
    const float* nbrr  = (const float*)d_in[2];   // [512,64,32,64]
    const float* uemb  = (const float*)d_in[3];   // [512,64]
    const float* W     = (const float*)d_in[4];   // [64,64]
    float*       out   = (float*)d_out;           // [512,64,64]

    const int pairs  = 512 * 64;                  // 32768
    const int blocks = pairs / TILE;              // 2048
    Aggregator_83691732730318_kernel<<<blocks, 512, 0, stream>>>(
        selfv, nbrv, nbrr, uemb, W, out);
}